// ChebshevGNN_66898410603231
// MI455X (gfx1250) — compile-verified
//
#include <hip/hip_runtime.h>
#include <hip/hip_bf16.h>

// Problem constants (match reference)
#define BB 8
#define NN 4096
#define EE 65536
#define FF 128
#define MM 8
#define KP1 4   // K+1 Chebyshev terms

typedef __attribute__((ext_vector_type(2))) float v2f;
typedef __attribute__((ext_vector_type(4))) float f4;
typedef __attribute__((ext_vector_type(8))) float v8f;

// ---------------------------------------------------------------------------
// Guaranteed hardware FP32 atomic add to global memory (no CAS fallback).
// ---------------------------------------------------------------------------
__device__ __forceinline__ void gatomic_add_f32(float* p, float v) {
    asm volatile("global_atomic_add_f32 %0, %1, off" :: "v"(p), "v"(v) : "memory");
}

// ---------------------------------------------------------------------------
// y = a * src   (or 0 if src == nullptr). Vectorized 16B init of recursion
// buffers; folds the "- x_{k-2}" term of the Chebyshev recursion into init.
// ---------------------------------------------------------------------------
__global__ void init_scaled(f4* __restrict__ y, const f4* __restrict__ s,
                            float a, int n4) {
    int i = blockIdx.x * blockDim.x + threadIdx.x;
    if (i < n4) {
        f4 v;
        if (s) v = s[i] * a;
        else   v = (f4){0.f, 0.f, 0.f, 0.f};
        y[i] = v;
    }
}

// ---------------------------------------------------------------------------
// SpMM scatter pass: y[b,dst,:] += scale * w[b,e] * x[b,src,:]
// One wave (32 lanes) covers a full F=128 row as float4; 4 edges per wave
// with global_prefetch of gathered rows ahead of use.
// ---------------------------------------------------------------------------
__global__ void __launch_bounds__(256) spmm_scatter(
        const float* __restrict__ x, float* __restrict__ y,
        const float* __restrict__ ew, const int* __restrict__ esrc,
        const int* __restrict__ edst, float scale) {
    const int lane = threadIdx.x & 31;
    const int wave = threadIdx.x >> 5;
    const long gw = (long)blockIdx.x * 8 + wave;   // global wave id
    const long e0 = gw * 4;                        // 4 edges per wave
    const int b     = (int)(e0 / EE);              // EE % 4 == 0 -> same batch
    const int ebase = (int)(e0 % EE);

    const float* xb = x + (size_t)b * NN * FF;
    float*       yb = y + (size_t)b * NN * FF;

    int src[4], dst[4]; float w[4];
#pragma unroll
    for (int i = 0; i < 4; ++i) {
        src[i] = esrc[ebase + i];
        dst[i] = edst[ebase + i];
        w[i]   = ew[(size_t)b * EE + ebase + i] * scale;
    }
#pragma unroll
    for (int i = 0; i < 4; ++i)
        __builtin_prefetch(xb + (size_t)src[i] * FF + lane * 4, 0, 0);

#pragma unroll
    for (int i = 0; i < 4; ++i) {
        const f4* row = (const f4*)(xb + (size_t)src[i] * FF);
        f4 v = row[lane] * w[i];
        float* o = yb + (size_t)dst[i] * FF + lane * 4;
        gatomic_add_f32(o + 0, v.x);
        gatomic_add_f32(o + 1, v.y);
        gatomic_add_f32(o + 2, v.z);
        gatomic_add_f32(o + 3, v.w);
    }
}

// ---------------------------------------------------------------------------
// Fused einsum('bnkf,mkf->bnmf') + bias + ReLU via V_WMMA_F32_16X16X4_F32.
// Contraction depth is exactly 4 == WMMA K. Per feature channel f:
//   A[row,k] = cheb_k[b, n0+row, f]      (16x4)
//   B[k,m]   = weight[m, k, f]           (4x16, 8 valid cols)
//   D[row,m] -> relu(D + bias[m*F+f]) -> out[b, n0+row, m*F+f]
// A layout: lane holds row = lane%16; VGPR0 = K (0|2 by lane half), VGPR1=K+1.
// D layout: row = vgpr + 8*(lane/16), col = lane%16.
// D is staged in LDS (pad 17, conflict-free) so global stores are coalesced
// 64B segments with f varying across lanes.
// ---------------------------------------------------------------------------
__global__ void __launch_bounds__(128) cheb_combine(
        const float* __restrict__ c0, const float* __restrict__ c1,
        const float* __restrict__ c2, const float* __restrict__ c3,
        const float* __restrict__ wgt,   // (M, 4, F)
        const float* __restrict__ bias,  // (M*F)
        float* __restrict__ out) {       // (B, N, M*F)
    __shared__ float stage[4][16 * 8 * 17];   // per-wave 16-f chunk staging

    const int lane = threadIdx.x & 31;
    const int wv   = threadIdx.x >> 5;
    const int bI   = blockIdx.x / (NN / 16);
    const int n0   = (blockIdx.x % (NN / 16)) * 16;

    const int row = lane & 15;     // A-matrix row / D row base
    const int kh  = lane >> 4;     // lane half: K offset 0 or 2
    const int m   = lane & 15;     // B-matrix column (filter index)
    const bool mok = (m < MM);
    const int msafe = mok ? m : 0;

    const float* ca = kh ? c2 : c0;              // A VGPR0 source (K = 2*kh)
    const float* cb = kh ? c3 : c1;              // A VGPR1 source (K = 2*kh+1)
    const size_t abase = ((size_t)bI * NN + n0 + row) * FF;
    const int wb0 = (msafe * KP1 + 2 * kh) * FF;
    const int wb1 = (msafe * KP1 + 2 * kh + 1) * FF;
    const float bm = mok ? 1.f : 0.f;

    float* stg = stage[wv];

    for (int c = 0; c < 2; ++c) {                // two 16-f chunks per wave
        const int f0 = wv * 32 + c * 16;

        // Wide B128 loads: 16 consecutive f per lane for every operand.
        f4 A0[4], A1[4], W0[4], W1[4], BS[4];
        const f4* pA0 = (const f4*)(ca + abase + f0);
        const f4* pA1 = (const f4*)(cb + abase + f0);
        const f4* pW0 = (const f4*)(wgt + wb0 + f0);
        const f4* pW1 = (const f4*)(wgt + wb1 + f0);
        const f4* pBS = (const f4*)(bias + msafe * FF + f0);
#pragma unroll
        for (int q = 0; q < 4; ++q) {
            A0[q] = pA0[q]; A1[q] = pA1[q];
            W0[q] = pW0[q] * bm; W1[q] = pW1[q] * bm;
            BS[q] = pBS[q];
        }
        const float* a0 = (const float*)A0;
        const float* a1 = (const float*)A1;
        const float* w0 = (const float*)W0;
        const float* w1 = (const float*)W1;
        const float* bs = (const float*)BS;

#pragma unroll
        for (int fi = 0; fi < 16; ++fi) {
            v2f A; A.x = a0[fi]; A.y = a1[fi];
            v2f Bv; Bv.x = w0[fi]; Bv.y = w1[fi];
            v8f C = {};
            // D = A(16x4) x B(4x16) + 0   -> v_wmma_f32_16x16x4_f32
            C = __builtin_amdgcn_wmma_f32_16x16x4_f32(
                    false, A, false, Bv, (short)0, C, false, false);
            if (mok) {
                const float bv = bs[fi];
#pragma unroll
                for (int r = 0; r < 8; ++r) {
                    const int orow = r + 8 * kh;
                    float v = C[r] + bv;
                    v = v > 0.f ? v : 0.f;
                    stg[(orow * 8 + m) * 17 + fi] = v;   // conflict-free pad
                }
            }
        }

        // Coalesced write-back: 2 (row,m) combos per iter, f across lanes.
#pragma unroll 4
        for (int j = 0; j < 64; ++j) {
            const int combo = j * 2 + kh;       // 0..127 = row*8+m
            const int orow  = combo >> 3;
            const int om    = combo & 7;
            const int fi    = lane & 15;
            const float v = stg[combo * 17 + fi];
            out[((size_t)bI * NN + n0 + orow) * (MM * FF) + om * FF + f0 + fi] = v;
        }
    }
}

// ---------------------------------------------------------------------------
// Launch: x1 = L x0 ; x2 = 2 L x1 - x0 ; x3 = 2 L x2 - x1 ; fused einsum.
// The "- x_{k-2}" is folded into buffer init, so each step = init + scatter.
// ---------------------------------------------------------------------------
extern "C" void kernel_launch(void* const* d_in, const int* in_sizes, int n_in,
                              void* d_out, int out_size, void* d_ws, size_t ws_size,
                              hipStream_t stream) {
    const float* x    = (const float*)d_in[0];   // (B,N,F)
    const float* ew   = (const float*)d_in[1];   // (B,E)
    const float* wgt  = (const float*)d_in[2];   // (M,K+1,F)
    const float* bias = (const float*)d_in[3];   // (M*F)
    const int*   esrc = (const int*)d_in[4];     // (E)
    const int*   edst = (const int*)d_in[5];     // (E)
    float* out = (float*)d_out;                  // (B,N,M*F)

    const size_t BNF = (size_t)BB * NN * FF;     // 4,194,304 floats (16 MB)
    float* y1 = (float*)d_ws;
    float* y2 = y1 + BNF;
    float* y3 = y2 + BNF;

    const int n4 = (int)(BNF / 4);
    const int initBlocks = (n4 + 255) / 256;
    const int spmmBlocks = (BB * EE) / (4 * 8); // 4 edges/wave, 8 waves/block
    const int combBlocks = BB * (NN / 16);

    // x1 = L x0
    init_scaled<<<initBlocks, 256, 0, stream>>>((f4*)y1, nullptr, 0.f, n4);
    spmm_scatter<<<spmmBlocks, 256, 0, stream>>>(x, y1, ew, esrc, edst, 1.f);
    // x2 = 2 L x1 - x0
    init_scaled<<<initBlocks, 256, 0, stream>>>((f4*)y2, (const f4*)x, -1.f, n4);
    spmm_scatter<<<spmmBlocks, 256, 0, stream>>>(y1, y2, ew, esrc, edst, 2.f);
    // x3 = 2 L x2 - x1
    init_scaled<<<initBlocks, 256, 0, stream>>>((f4*)y3, (const f4*)y1, -1.f, n4);
    spmm_scatter<<<spmmBlocks, 256, 0, stream>>>(y2, y3, ew, esrc, edst, 2.f);
    // out = relu(einsum(cheb, weight) + bias)
    cheb_combine<<<combBlocks, 128, 0, stream>>>(x, y1, y2, y3, wgt, bias, out);
}